// SAGEConv_74526272520731
// MI455X (gfx1250) — compile-verified
//
#include <hip/hip_runtime.h>

#define N_NODES 100000
#define F_IN    64
#define F_TOT   128
#define F_OUT   64
#define LDS_ST  130   // row stride in floats: keeps 8B alignment for ds_load_b64

typedef __attribute__((ext_vector_type(2))) float v2f;
typedef __attribute__((ext_vector_type(8))) float v8f;

// ---------------- kernel 1: zero accumulators (summed[N*64] + deg[N]) -------
__global__ void sage_zero(float* __restrict__ ws, int n) {
    int i = blockIdx.x * blockDim.x + threadIdx.x;
    if (i < n) ws[i] = 0.0f;
}

// ---------------- kernel 2: edge gather + scatter-add -----------------------
// 16 threads per edge; each thread moves a float4 of the 64-wide feature row.
__global__ void sage_edges(const float* __restrict__ h_src,
                           const int*   __restrict__ src,
                           const int*   __restrict__ dst,
                           float*       __restrict__ summed,
                           float*       __restrict__ deg,
                           int n_edges) {
    int t = blockIdx.x * blockDim.x + threadIdx.x;
    int e = t >> 4;
    if (e >= n_edges) return;
    int q = t & 15;
    int s = src[e];
    int d = dst[e];
    const float4 v = ((const float4*)(h_src + (size_t)s * F_IN))[q];
    float* drow = summed + (size_t)d * F_IN + q * 4;
    atomicAdd(drow + 0, v.x);
    atomicAdd(drow + 1, v.y);
    atomicAdd(drow + 2, v.z);
    atomicAdd(drow + 3, v.w);
    if (q == 0) atomicAdd(deg + d, 1.0f);
}

// ---------------- kernel 3: X = [h_dst | summed/deg], out = X*W^T + b -------
// 64 threads = 2 waves. Block tile: 32 nodes. Wave tile: 16 nodes x 64 out.
// K=128 reduced via 32 chained v_wmma_f32_16x16x4_f32 per output tile.
__global__ void __launch_bounds__(64) sage_gemm(
        const float* __restrict__ h_dst,
        const float* __restrict__ summed,
        const float* __restrict__ deg,
        const float* __restrict__ weight,   // [64][128] row-major (out x in)
        const float* __restrict__ bias,     // [64]
        float*       __restrict__ out) {    // [N][64]
    __shared__ float Xs[32 * LDS_ST];       // 32 nodes x 128 (padded)
    __shared__ float Wsh[64 * LDS_ST];      // 64 out   x 128 (padded)

    const int tid       = threadIdx.x;      // 0..63
    const int node_base = blockIdx.x * 32;

    // Stage weight: 64 rows x 128 cols, 64 threads -> 2 cols per row.
    for (int r = 0; r < 64; ++r) {
        Wsh[r * LDS_ST + tid]      = weight[r * F_TOT + tid];
        Wsh[r * LDS_ST + 64 + tid] = weight[r * F_TOT + 64 + tid];
    }
    // Stage X: 32 rows; col tid from h_dst, col 64+tid = neighbor mean.
    for (int r = 0; r < 32; ++r) {
        int node = node_base + r;
        int cn   = node < N_NODES ? node : (N_NODES - 1);
        float dg = deg[cn];
        float hn = summed[(size_t)cn * F_IN + tid] / fmaxf(dg, 1.0f);
        Xs[r * LDS_ST + tid]      = h_dst[(size_t)cn * F_IN + tid];
        Xs[r * LDS_ST + 64 + tid] = hn;
    }
    __syncthreads();

    const int wave = tid >> 5;              // 0..1
    const int lane = tid & 31;
    const int nrow = lane & 15;             // A: M row / B: N col within tile
    const int hi   = lane >> 4;             // half-wave -> K sub-offset (0 or 2)

    v8f c0 = {}, c1 = {}, c2 = {}, c3 = {};

    const float* Xrow = &Xs[(wave * 16 + nrow) * LDS_ST + 2 * hi];
    const float* W0   = &Wsh[( 0 + nrow) * LDS_ST + 2 * hi];
    const float* W1   = &Wsh[(16 + nrow) * LDS_ST + 2 * hi];
    const float* W2   = &Wsh[(32 + nrow) * LDS_ST + 2 * hi];
    const float* W3   = &Wsh[(48 + nrow) * LDS_ST + 2 * hi];

    #pragma unroll
    for (int k = 0; k < F_TOT; k += 4) {
        v2f a  = *(const v2f*)(Xrow + k);   // A 16x4 f32 fragment
        v2f b0 = *(const v2f*)(W0 + k);     // B 4x16 f32 fragments (4 o-tiles)
        v2f b1 = *(const v2f*)(W1 + k);
        v2f b2 = *(const v2f*)(W2 + k);
        v2f b3 = *(const v2f*)(W3 + k);
        c0 = __builtin_amdgcn_wmma_f32_16x16x4_f32(false, a, false, b0, (short)0, c0, false, false);
        c1 = __builtin_amdgcn_wmma_f32_16x16x4_f32(false, a, false, b1, (short)0, c1, false, false);
        c2 = __builtin_amdgcn_wmma_f32_16x16x4_f32(false, a, false, b2, (short)0, c2, false, false);
        c3 = __builtin_amdgcn_wmma_f32_16x16x4_f32(false, a, false, b3, (short)0, c3, false, false);
    }

    // C/D layout: VGPR j -> M = j + 8*hi, N = lane%16.
    #pragma unroll
    for (int j = 0; j < 8; ++j) {
        int m    = j + 8 * hi;
        int node = node_base + wave * 16 + m;
        if (node < N_NODES) {
            float* orow = out + (size_t)node * F_OUT;
            orow[ 0 + nrow] = c0[j] + bias[ 0 + nrow];
            orow[16 + nrow] = c1[j] + bias[16 + nrow];
            orow[32 + nrow] = c2[j] + bias[32 + nrow];
            orow[48 + nrow] = c3[j] + bias[48 + nrow];
        }
    }
}

extern "C" void kernel_launch(void* const* d_in, const int* in_sizes, int n_in,
                              void* d_out, int out_size, void* d_ws, size_t ws_size,
                              hipStream_t stream) {
    const float* h_src  = (const float*)d_in[0];
    const float* h_dst  = (const float*)d_in[1];
    const int*   src    = (const int*)d_in[2];
    const int*   dst    = (const int*)d_in[3];
    const float* weight = (const float*)d_in[4];
    const float* bias   = (const float*)d_in[5];
    float*       outp   = (float*)d_out;

    const int n_edges = in_sizes[2];

    float* summed = (float*)d_ws;                      // N*64 floats
    float* degp   = summed + (size_t)N_NODES * F_IN;   // N floats

    // 1) zero accumulators
    int nz = N_NODES * F_IN + N_NODES;
    sage_zero<<<(nz + 255) / 256, 256, 0, stream>>>((float*)d_ws, nz);

    // 2) edge scatter (16 threads per edge)
    long long et = (long long)n_edges * 16;
    sage_edges<<<(int)((et + 255) / 256), 256, 0, stream>>>(
        h_src, src, dst, summed, degp, n_edges);

    // 3) fused mean/concat/linear via f32 WMMA
    sage_gemm<<<(N_NODES + 31) / 32, 64, 0, stream>>>(
        h_dst, summed, degp, weight, bias, outp);
}